// GotenNet_52905407152733
// MI455X (gfx1250) — compile-verified
//
#include <hip/hip_runtime.h>
#include <math.h>

#define NN 10000      // nodes
#define NE 160000     // edges
#define DD 128        // node/edge dim
#define RBF 64
#define HEADS 8
#define CUT 5.0f
#define TK_MAX 256    // max K of any GEMM here

typedef __attribute__((ext_vector_type(2))) float v2f;
typedef __attribute__((ext_vector_type(8))) float v8f;

__device__ __forceinline__ float silu_f(float x) { return x / (1.0f + __expf(-x)); }

__device__ inline void atomicMaxF(float* addr, float val) {
  unsigned int* ua = (unsigned int*)addr;
  unsigned int old = *ua;
  while (__uint_as_float(old) < val) {
    unsigned int assumed = old;
    old = atomicCAS(ua, assumed, __float_as_uint(val));
    if (old == assumed) break;
  }
}

// ---- CDNA5 async global->LDS staging (guarded; fallback = load + ds_store) ----
#if defined(__has_builtin)
#if __has_builtin(__builtin_amdgcn_global_load_async_to_lds_b128) && \
    __has_builtin(__builtin_amdgcn_s_wait_asynccnt)
#define HAVE_ASYNC_LDS 1
#endif
#endif
#ifndef HAVE_ASYNC_LDS
#define HAVE_ASYNC_LDS 0
#endif

// builtin expects pointers to int4 vectors (per hipcc diagnostic):
//   global: __attribute__((vector_size(16))) int __device__ *
//   lds:    __attribute__((vector_size(16))) int __shared__ *
typedef int v4i_t __attribute__((vector_size(4 * sizeof(int))));
typedef __attribute__((address_space(1))) v4i_t gbl_v4i_t;
typedef __attribute__((address_space(3))) v4i_t lds_v4i_t;

__device__ __forceinline__ void g2l_b128(const float* g, float* l) {
#if HAVE_ASYNC_LDS
  // generic LDS pointer: low 32 bits are the LDS byte offset (ISA 10.2)
  __builtin_amdgcn_global_load_async_to_lds_b128(
      (gbl_v4i_t*)(unsigned long long)g,
      (lds_v4i_t*)(unsigned int)(unsigned long long)l, 0, 0);
#else
  *(float4*)l = *(const float4*)g;   // global_load_b128 + ds_store_b128
#endif
}
__device__ __forceinline__ void g2l_commit() {
#if HAVE_ASYNC_LDS
  __builtin_amdgcn_s_wait_asynccnt(0);
#endif
}

// ---------------------------------------------------------------------------
// fp32 WMMA GEMM: C = act(A @ W + bias)
// Row remap (shared by A and C): phys_off(r) = base + (r/div)*blk + (r%div)*row
// Block = 128 threads (4 waves) -> 16x64 C tile; A tile (16xK) staged in LDS
// via GLOBAL_LOAD_ASYNC_TO_LDS_B128, each wave runs V_WMMA_F32_16X16X4_F32.
// Requires M%16==0, N%64==0, K%32==0 (true for every call site here).
// ---------------------------------------------------------------------------
__global__ void gemm16_wmma(const float* __restrict__ A, const float* __restrict__ W,
                            const float* __restrict__ bias, float* __restrict__ C,
                            int N, int K,
                            int adiv, long ablk, long arow, long abase,
                            long cblk, long crow, long cbase, int act)
{
  __shared__ float As[16 * TK_MAX];
  const int tm = blockIdx.x, tn = blockIdx.y;
  const int tid = threadIdx.x;                 // 0..127
  // ---- stage A tile (16 rows x K) into LDS in 16B chunks ----
  const int nchunk = (16 * K) >> 2;            // multiple of 128
  for (int ci = tid; ci < nchunk; ci += 128) {
    int fl = ci << 2;
    int row = fl / K, kp = fl % K;             // kp multiple of 4 (K%32==0)
    int rA = tm * 16 + row;
    const float* g = A + abase + (long)(rA / adiv) * ablk + (long)(rA % adiv) * arow + kp;
    g2l_b128(g, &As[fl]);
  }
  g2l_commit();
  __syncthreads();
  // ---- per-wave 16x16 tile ----
  const int lane = tid & 31;
  const int wv = tid >> 5;                     // wave id 0..3
  const int half = lane >> 4;                  // 0: K=k,k+1 ; 1: K=k+2,k+3
  const int l15 = lane & 15;
  const int col = tn * 64 + wv * 16 + l15;
  const float* __restrict__ wp = W + (long)(2 * half) * N + col;
  const float* ar = &As[l15 * K + 2 * half];   // ds_load_b64 per step
  v8f acc = {};
  for (int k = 0; k < K; k += 4) {
    v2f a; a.x = ar[k];           a.y = ar[k + 1];
    v2f b; b.x = wp[(long)k * N]; b.y = wp[(long)(k + 1) * N];
    acc = __builtin_amdgcn_wmma_f32_16x16x4_f32(false, a, false, b, (short)0, acc,
                                                false, false);
  }
  float bv = bias ? bias[col] : 0.0f;
#pragma unroll
  for (int r = 0; r < 8; ++r) {                // C VGPR r: lanes0-15 M=r, 16-31 M=r+8
    int rowC = tm * 16 + half * 8 + r;
    long co = cbase + (long)(rowC / adiv) * cblk + (long)(rowC % adiv) * crow + col;
    float v = acc[r] + bv;
    if (act) v = silu_f(v);
    C[co] = v;
  }
}

// ---------------------------------------------------------------------------
// Elementwise / scatter kernels
// ---------------------------------------------------------------------------
__global__ void k_zero(float* p, long n) {
  long i = blockIdx.x * (long)blockDim.x + threadIdx.x;
  if (i < n) p[i] = 0.0f;
}
__global__ void k_fillv(float* p, float v, long n) {
  long i = blockIdx.x * (long)blockDim.x + threadIdx.x;
  if (i < n) p[i] = v;
}
__global__ void k_add(float* d, const float* a, long n) {
  long i = blockIdx.x * (long)blockDim.x + threadIdx.x;
  if (i < n) d[i] += a[i];
}
__global__ void k_muladd(float* d, const float* a, const float* b, long n) {
  long i = blockIdx.x * (long)blockDim.x + threadIdx.x;
  if (i < n) d[i] += a[i] * b[i];
}

// per-edge geometry: r0, unit vec, l=2 spherical harmonics, cutoff, RBF
__global__ void k_geom(const float* __restrict__ xe, const float* __restrict__ ctr,
                       const float* __restrict__ gam, float* r0, float* r1,
                       float* sh, float* cc, float* rbf0)
{
  int e = blockIdx.x * blockDim.x + threadIdx.x;
  if (e >= NE) return;
  float x = xe[e * 3], y = xe[e * 3 + 1], z = xe[e * 3 + 2];
  float r = sqrtf(x * x + y * y + z * z);
  r0[e] = r;
  float inv = 1.0f / r;
  float ux = x * inv, uy = y * inv, uz = z * inv;
  r1[e * 3] = ux; r1[e * 3 + 1] = uy; r1[e * 3 + 2] = uz;
  const float c15 = 3.8729833462074170f;   // sqrt(15)
  const float s5  = 2.2360679774997896f;   // sqrt(5)
  const float is5 = 1.0f / s5;
  sh[e * 5 + 0] = c15 * ux * uz * is5;
  sh[e * 5 + 1] = c15 * ux * uy * is5;
  sh[e * 5 + 2] = (s5 * 0.5f) * (3.0f * uy * uy - 1.0f) * is5;
  sh[e * 5 + 3] = c15 * uy * uz * is5;
  sh[e * 5 + 4] = (c15 * 0.5f) * (uz * uz - ux * ux) * is5;
  cc[e] = (r < CUT) ? 0.5f * (cosf(3.14159265358979323846f * r / CUT) + 1.0f) : 0.0f;
  for (int j = 0; j < RBF; ++j) {
    float d = r - ctr[j];
    rbf0[(long)e * RBF + j] = __expf(-gam[j] * d * d);
  }
}

// m[n_i] += a_nbr[z[n_j]] * ndp * cc      (grid: NE blocks x 128 threads)
__global__ void k_scatter_m(const float* __restrict__ ndp, const float* __restrict__ cc,
                            const int* __restrict__ xn, const int* __restrict__ nj,
                            const int* __restrict__ ni, const float* __restrict__ a_nbr,
                            float* __restrict__ m)
{
  int e = blockIdx.x, c = threadIdx.x;
  int j = nj[e], i = ni[e];
  float v = a_nbr[(long)xn[j] * DD + c] * ndp[(long)e * DD + c] * cc[e];
  atomicAdd(&m[(long)i * DD + c], v);
}

__global__ void k_build_hin(const int* __restrict__ xn, const float* __restrict__ a_na,
                            const float* __restrict__ m, float* __restrict__ hin)
{
  int n = blockIdx.x, c = threadIdx.x;
  hin[(long)n * 256 + c] = (c < DD) ? a_na[(long)xn[n] * DD + c]
                                    : m[(long)n * DD + (c - DD)];
}

// per-row layernorm (optionally followed by SiLU); in-place safe
__global__ void k_layernorm(const float* __restrict__ in, float* __restrict__ out,
                            const float* __restrict__ g, const float* __restrict__ b,
                            int D, long rows, int act)
{
  long r = blockIdx.x * (long)blockDim.x + threadIdx.x;
  if (r >= rows) return;
  const float* x = in + r * D;
  float mean = 0.0f;
  for (int c = 0; c < D; ++c) mean += x[c];
  mean /= (float)D;
  float var = 0.0f;
  for (int c = 0; c < D; ++c) { float d = x[c] - mean; var += d * d; }
  var /= (float)D;
  float invs = rsqrtf(var + 1e-5f);
  float* o = out + r * D;
  for (int c = 0; c < D; ++c) {
    float v = (x[c] - mean) * invs * g[c] + b[c];
    if (act) v = silu_f(v);
    o[c] = v;
  }
}

// t_ij = (h[n_i] + h[n_j]) * erp          (grid: NE x 128)
__global__ void k_tij(const float* __restrict__ h, const float* __restrict__ erp,
                      const int* __restrict__ nj, const int* __restrict__ ni,
                      float* __restrict__ tij)
{
  int e = blockIdx.x, c = threadIdx.x;
  tij[(long)e * DD + c] =
      (h[(long)ni[e] * DD + c] + h[(long)nj[e] * DD + c]) * erp[(long)e * DD + c];
}

__global__ void k_attn_score(const float* __restrict__ q, const float* __restrict__ kk,
                             const float* __restrict__ re, const int* __restrict__ nj,
                             const int* __restrict__ ni, float* __restrict__ aexp,
                             float* __restrict__ mx, long total)
{
  long t = blockIdx.x * (long)blockDim.x + threadIdx.x;
  if (t >= total) return;
  long e = t >> 3; int hd = (int)(t & 7);
  int i = ni[e], j = nj[e];
  const float* qp = q  + (long)i * DD + hd * 16;
  const float* kp = kk + (long)j * DD + hd * 16;
  const float* rp = re + e * (long)DD + hd * 16;
  float s = 0.0f;
#pragma unroll
  for (int c = 0; c < 16; ++c) s += qp[c] * kp[c] * rp[c];
  aexp[t] = s;
  atomicMaxF(&mx[(long)i * HEADS + hd], s);
}

__global__ void k_mx_fix(float* mx, long n) {
  long t = blockIdx.x * (long)blockDim.x + threadIdx.x;
  if (t >= n) return;
  float v = mx[t];
  if (!(fabsf(v) <= 3.402823466e+38f)) mx[t] = 0.0f;  // non-finite -> 0
}

__global__ void k_deg(const int* __restrict__ ni, float* __restrict__ deg) {
  int e = blockIdx.x * blockDim.x + threadIdx.x;
  if (e >= NE) return;
  atomicAdd(&deg[ni[e]], 1.0f);
}

__global__ void k_attn_exp(float* __restrict__ aexp, const float* __restrict__ mx,
                           float* __restrict__ ssum, const int* __restrict__ ni, long total)
{
  long t = blockIdx.x * (long)blockDim.x + threadIdx.x;
  if (t >= total) return;
  long e = t >> 3; int hd = (int)(t & 7);
  int i = ni[e];
  float ev = __expf(aexp[t] - mx[(long)i * HEADS + hd]);
  aexp[t] = ev;
  atomicAdd(&ssum[(long)i * HEADS + hd], ev);
}

__global__ void k_attn_fin(float* __restrict__ aexp, const float* __restrict__ ssum,
                           const float* __restrict__ deg, const int* __restrict__ ni,
                           float inv_sqrt_dv, long total)
{
  long t = blockIdx.x * (long)blockDim.x + threadIdx.x;
  if (t >= total) return;
  long e = t >> 3; int hd = (int)(t & 7);
  int i = ni[e];
  aexp[t] = aexp[t] / (ssum[(long)i * HEADS + hd] + 1e-16f) * sqrtf(deg[i]) * inv_sqrt_dv;
}

// o = attn*v[n_j] + o*sv[n_j]*cc         (grid: NE blocks x outf threads)
__global__ void k_o_combine(float* __restrict__ o, const float* __restrict__ aexp,
                            const float* __restrict__ v, const float* __restrict__ sv,
                            const float* __restrict__ cc, const int* __restrict__ nj,
                            int outf, int dv)
{
  long e = blockIdx.x; int c = threadIdx.x;
  int j = nj[e];
  int hd = c / dv;
  long idx = e * (long)outf + c;
  o[idx] = aexp[e * HEADS + hd] * v[(long)j * outf + c]
         + o[idx] * sv[(long)j * outf + c] * cc[e];
}

// embedding: X[n_i, d] += o_split[deg] * dir   (grid: NE x 128)
__global__ void k_scat_emb(const float* __restrict__ o, const float* __restrict__ r1,
                           const float* __restrict__ sh, const int* __restrict__ ni,
                           float* __restrict__ Xall)
{
  int e = blockIdx.x, c = threadIdx.x;
  int i = ni[e];
  float p0 = o[(long)e * 256 + c], p1 = o[(long)e * 256 + 128 + c];
#pragma unroll
  for (int d = 0; d < 3; ++d)
    atomicAdd(&Xall[(long)i * 1024 + d * 128 + c], p0 * r1[e * 3 + d]);
#pragma unroll
  for (int d = 0; d < 5; ++d)
    atomicAdd(&Xall[(long)i * 1024 + (3 + d) * 128 + c], p1 * sh[e * 5 + d]);
}

// w[e,c] = sum_r EQ[n_i*8+r, c] * EK[n_j*8+r, c]
__global__ void k_htr_w(const float* __restrict__ EQ, const float* __restrict__ EK,
                        const int* __restrict__ nj, const int* __restrict__ ni,
                        float* __restrict__ wbuf, long total)
{
  long t = blockIdx.x * (long)blockDim.x + threadIdx.x;
  if (t >= total) return;
  long e = t >> 6; int c = (int)(t & 63);
  int i = ni[e], j = nj[e];
  float s = 0.0f;
#pragma unroll
  for (int r = 0; r < 8; ++r)
    s += EQ[((long)i * 8 + r) * 64 + c] * EK[((long)j * 8 + r) * 64 + c];
  wbuf[t] = s;
}

// gata scatter: h += parts[0]; dX += od*dir + ot*X[n_j]    (grid: NE x 128)
__global__ void k_scat_gata(const float* __restrict__ o, const float* __restrict__ r1,
                            const float* __restrict__ sh, const float* __restrict__ Xold,
                            const int* __restrict__ nj, const int* __restrict__ ni,
                            float* __restrict__ hacc, float* __restrict__ dX)
{
  int e = blockIdx.x, c = threadIdx.x;
  int i = ni[e], j = nj[e];
  const float* oe = o + (long)e * 640;
  atomicAdd(&hacc[(long)i * 128 + c], oe[c]);
  float p1 = oe[128 + c], p2 = oe[256 + c], p3 = oe[384 + c], p4 = oe[512 + c];
  const float* Xj = Xold + (long)j * 1024 + c;
  __builtin_prefetch(Xj, 0, 0);  // global_prefetch_b8
#pragma unroll
  for (int d = 0; d < 3; ++d)
    atomicAdd(&dX[(long)i * 1024 + d * 128 + c], p1 * r1[e * 3 + d] + p3 * Xj[d * 128]);
#pragma unroll
  for (int d = 0; d < 5; ++d)
    atomicAdd(&dX[(long)i * 1024 + (3 + d) * 128 + c],
              p2 * sh[e * 5 + d] + p4 * Xj[(3 + d) * 128]);
}

// cat = [ ||Xvu||_2 over d , h ]          (grid: NN x 128)
__global__ void k_l2cat(const float* __restrict__ Xvu, const float* __restrict__ h,
                        float* __restrict__ cat)
{
  int n = blockIdx.x, c = threadIdx.x;
  float s = 0.0f;
#pragma unroll
  for (int d = 0; d < 8; ++d) { float v = Xvu[(long)n * 1024 + d * 128 + c]; s += v * v; }
  cat[(long)n * 256 + c] = sqrtf(s + 1e-8f);
  cat[(long)n * 256 + 128 + c] = h[(long)n * 128 + c];
}

// h += m1; Xall += m2 * Xvu               (grid: NN x 128)
__global__ void k_eqff(const float* __restrict__ mo, const float* __restrict__ Xvu,
                       float* __restrict__ h, float* __restrict__ Xall)
{
  int n = blockIdx.x, c = threadIdx.x;
  h[(long)n * 128 + c] += mo[(long)n * 256 + c];
  float m2 = mo[(long)n * 256 + 128 + c];
#pragma unroll
  for (int d = 0; d < 8; ++d)
    Xall[(long)n * 1024 + d * 128 + c] += m2 * Xvu[(long)n * 1024 + d * 128 + c];
}

// decoder tail: out[batch[n]] += dot(d1[n], w2) + b2
__global__ void k_dec(const float* __restrict__ d1, const float* __restrict__ w2,
                      const float* __restrict__ b2, const int* __restrict__ batch,
                      float* __restrict__ out)
{
  int n = blockIdx.x * blockDim.x + threadIdx.x;
  if (n >= NN) return;
  float s = 0.0f;
  for (int c = 0; c < 128; ++c) s += d1[(long)n * 128 + c] * w2[c];
  s += b2[0];
  atomicAdd(&out[batch[n]], s);
}

// ---------------------------------------------------------------------------
// Host orchestration
// ---------------------------------------------------------------------------
struct SeaPtr {
  const float *v1b, *v1w, *v2b, *v2w;   // mlp_v.l1{b,w}, mlp_v.l2{b,w}
  const float *reb, *rew;               // w_re{b,w}
  const float *wkb, *wkw, *wqb, *wqw;   // wk{b,w}, wq{b,w}
};

extern "C" void kernel_launch(void* const* d_in, const int* in_sizes, int n_in,
                              void* d_out, int out_size, void* d_ws, size_t ws_size,
                              hipStream_t stream)
{
  (void)in_sizes; (void)n_in; (void)out_size; (void)ws_size;
  const int*   x_n   = (const int*)d_in[0];
  const float* x_e   = (const float*)d_in[1];
  const int*   eidx  = (const int*)d_in[2];
  const int*   n_j   = eidx;        // edge_index[0]
  const int*   n_i   = eidx + NE;   // edge_index[1]
  const int*   batch = (const int*)d_in[3];

  // ---- params: jax pytree flatten order (dict keys sorted, lists in order) ----
  int cur = 4;
  auto P = [&]() { return (const float*)d_in[cur++]; };
  // dec (mlp, keys l1<l2; lin keys b<w)
  const float *dec_l1_b = P(), *dec_l1_w = P(), *dec_l2_b = P(), *dec_l2_w = P();
  // emb (keys: a_na,a_nbr,ln,mlp_s,rbf,sea,seq_l1,seq_l2,seq_ln,w_erp,w_ndp,w_rs)
  const float *a_na = P(), *a_nbr = P();
  const float *eln_b = P(), *eln_g = P();
  const float *es_l1b = P(), *es_l1w = P(), *es_l2b = P(), *es_l2w = P(); // emb.mlp_s
  const float *rbf_c = P(), *rbf_g = P();
  SeaPtr esea = { P(), P(), P(), P(), P(), P(), P(), P(), P(), P() };     // emb.sea
  const float *sq1_b = P(), *sq1_w = P();   // seq_l1
  const float *sq2_b = P(), *sq2_w = P();   // seq_l2
  const float *sqln_b = P(), *sqln_g = P(); // seq_ln
  const float *erp_b = P(), *erp_w = P();   // w_erp
  const float *ndp_b = P(), *ndp_w = P();   // w_ndp
  const float *ers_b = P(), *ers_w = P();   // emb.w_rs
  struct LayerPtr {
    const float *mm_l1b,*mm_l1w,*mm_l2b,*mm_l2w,*mm_lnb,*mm_lng,*vu_w;       // eqff
    const float *mt_l1b,*mt_l1w,*mt_l2b,*mt_l2w;                              // htr.mlp_t
    const float *mw_l1b,*mw_l1w,*mw_l2b,*mw_l2w,*mw_lnb,*mw_lng;              // htr.mlp_w
    const float *vk0,*vk1,*vq;                                                // htr w_vk,w_vq
    const float *gln_b,*gln_g;                                                // gata.ln
    const float *ms_l1b,*ms_l1w,*ms_l2b,*ms_l2w;                              // gata.mlp_s
    SeaPtr sea;
    const float *rs_b,*rs_w;                                                  // gata.w_rs
  } L[2];
  for (int li = 0; li < 2; ++li) {   // layer dict: eqff < gata
    LayerPtr& l = L[li];
    l.mm_l1b=P(); l.mm_l1w=P(); l.mm_l2b=P(); l.mm_l2w=P(); l.mm_lnb=P(); l.mm_lng=P();
    l.vu_w=P();
    l.mt_l1b=P(); l.mt_l1w=P(); l.mt_l2b=P(); l.mt_l2w=P();
    l.mw_l1b=P(); l.mw_l1w=P(); l.mw_l2b=P(); l.mw_l2w=P(); l.mw_lnb=P(); l.mw_lng=P();
    l.vk0=P(); l.vk1=P(); l.vq=P();
    l.gln_b=P(); l.gln_g=P();
    l.ms_l1b=P(); l.ms_l1w=P(); l.ms_l2b=P(); l.ms_l2w=P();
    l.sea = { P(), P(), P(), P(), P(), P(), P(), P(), P(), P() };
    l.rs_b=P(); l.rs_w=P();
  }

  // ---- workspace layout ----
  char* wsp = (char*)d_ws;
  size_t off = 0;
  auto alloc = [&](long nf) -> float* {
    float* p = (float*)(wsp + off);
    off += (size_t)(((nf * 4 + 255) / 256) * 256);
    return p;
  };
  float *r0  = alloc(NE),       *r1 = alloc(3L*NE), *shb = alloc(5L*NE);
  float *ccb = alloc(NE),       *rbf0 = alloc(64L*NE);
  float *tij = alloc(128L*NE);
  float *EB1 = alloc(128L*NE),  *EB2 = alloc(128L*NE), *EB3 = alloc(128L*NE);
  float *obuf= alloc(640L*NE);  // also aliased below for htr scratch
  float *aexp= alloc(8L*NE);
  // htr scratch aliases obuf (timeline-disjoint: htr runs before w_rs GEMM fills obuf)
  float *EQb = obuf;
  float *EKb = obuf + (long)NN*8*64;
  float *wbuf= obuf + 2L*NN*8*64;
  float *w2  = wbuf + 64L*NE;
  // node buffers
  float *m   = alloc(128L*NN), *hin = alloc(256L*NN);
  float *t1  = alloc(128L*NN), *t2  = alloc(128L*NN);
  float *hA  = alloc(128L*NN), *hB  = alloc(128L*NN);
  float *qb  = alloc(128L*NN), *kb  = alloc(128L*NN);
  float *vtmp= alloc(128L*NN), *vb  = alloc(640L*NN);
  float *stmp= alloc(128L*NN), *sv  = alloc(640L*NN);
  float *mxb = alloc(8L*NN),   *ssum= alloc(8L*NN),  *degb = alloc(NN);
  float *Xall= alloc(1024L*NN),*dX  = alloc(1024L*NN),*Xvu = alloc(1024L*NN);
  float *catb= alloc(256L*NN), *catln = alloc(256L*NN);
  float *mh  = alloc(128L*NN), *mo  = alloc(256L*NN), *d1 = alloc(128L*NN);

  auto zero = [&](float* p, long n) {
    k_zero<<<(unsigned)((n + 255) / 256), 256, 0, stream>>>(p, n);
  };
  auto fill = [&](float* p, float v, long n) {
    k_fillv<<<(unsigned)((n + 255) / 256), 256, 0, stream>>>(p, v, n);
  };
  auto gemm = [&](const float* A, const float* W, const float* bias, float* C,
                  int M, int Ncol, int K, int act = 0, int adiv = 1,
                  long ablk = -1, long arow = 0, long abase = 0,
                  long cblk = -1, long crow = 0, long cbase = 0) {
    if (ablk < 0) ablk = K;
    if (cblk < 0) cblk = Ncol;
    gemm16_wmma<<<dim3(M / 16, Ncol / 64), 128, 0, stream>>>(
        A, W, bias, C, Ncol, K, adiv, ablk, arow, abase, cblk, crow, cbase, act);
  };
  auto lnorm = [&](const float* in, float* out, const float* g, const float* b,
                   int D, long rows, int act) {
    k_layernorm<<<(unsigned)((rows + 63) / 64), 64, 0, stream>>>(in, out, g, b, D, rows, act);
  };
  auto run_sea = [&](const SeaPtr& sp, const float* rs_b, const float* rs_w,
                     const float* ms1b, const float* ms1w,
                     const float* ms2b, const float* ms2w,
                     const float* hn, int outf, int dv) {
    gemm(hn, sp.wqw, sp.wqb, qb, NN, 128, 128);
    gemm(hn, sp.wkw, sp.wkb, kb, NN, 128, 128);
    gemm(hn, sp.v1w, sp.v1b, vtmp, NN, 128, 128, 1);
    gemm(vtmp, sp.v2w, sp.v2b, vb, NN, outf, 128);
    gemm(tij, sp.rew, sp.reb, EB3, NE, 128, 128, 1);         // re = silu(lin(t_ij))
    gemm(hn, ms1w, ms1b, stmp, NN, 128, 128, 1);             // mlp_s hidden
    gemm(stmp, ms2w, ms2b, sv, NN, outf, 128);
    gemm(tij, rs_w, rs_b, obuf, NE, outf, 128);              // rs -> o (in place combine)
    fill(mxb, -INFINITY, (long)NN * HEADS);
    zero(ssum, (long)NN * HEADS);
    zero(degb, NN);
    long tot = (long)NE * HEADS;
    k_attn_score<<<(unsigned)((tot + 255) / 256), 256, 0, stream>>>(qb, kb, EB3, n_j, n_i, aexp, mxb, tot);
    k_mx_fix<<<(unsigned)((NN * HEADS + 255) / 256), 256, 0, stream>>>(mxb, (long)NN * HEADS);
    k_deg<<<(NE + 255) / 256, 256, 0, stream>>>(n_i, degb);
    k_attn_exp<<<(unsigned)((tot + 255) / 256), 256, 0, stream>>>(aexp, mxb, ssum, n_i, tot);
    k_attn_fin<<<(unsigned)((tot + 255) / 256), 256, 0, stream>>>(aexp, ssum, degb, n_i,
                                                                  1.0f / sqrtf((float)dv), tot);
    k_o_combine<<<NE, outf, 0, stream>>>(obuf, aexp, vb, sv, ccb, n_j, outf, dv);
  };

  // ================== embedding ==================
  k_geom<<<(NE + 127) / 128, 128, 0, stream>>>(x_e, rbf_c, rbf_g, r0, r1, shb, ccb, rbf0);
  gemm(rbf0, ndp_w, ndp_b, EB1, NE, 128, 64);                // ndp
  zero(m, 128L * NN);
  k_scatter_m<<<NE, 128, 0, stream>>>(EB1, ccb, x_n, n_j, n_i, a_nbr, m);
  k_build_hin<<<NN, 256, 0, stream>>>(x_n, a_na, m, hin);
  gemm(hin, sq1_w, sq1_b, t1, NN, 128, 256);                 // seq_l1
  lnorm(t1, t2, sqln_g, sqln_b, 128, NN, 1);                 // act(LN(.))
  gemm(t2, sq2_w, sq2_b, hA, NN, 128, 128);                  // seq_l2 -> h (pre-LN)
  gemm(rbf0, erp_w, erp_b, EB2, NE, 128, 64);                // erp
  k_tij<<<NE, 128, 0, stream>>>(hA, EB2, n_j, n_i, tij);     // t_ij (uses pre-LN h)
  lnorm(hA, hB, eln_g, eln_b, 128, NN, 0);                   // h = LN(h)
  float* hcur = hB; float* hother = hA;
  run_sea(esea, ers_b, ers_w, es_l1b, es_l1w, es_l2b, es_l2w, hcur, 256, 32);
  zero(Xall, 1024L * NN);
  k_scat_emb<<<NE, 128, 0, stream>>>(obuf, r1, shb, n_i, Xall);

  // ================== gata + eqff layers ==================
  for (int li = 0; li < 2; ++li) {
    const LayerPtr& l = L[li];
    // hn = LN(h)
    lnorm(hcur, hother, l.gln_g, l.gln_b, 128, NN, 0);
    float* hn = hother;
    // ---- htr ----
    gemm(Xall, l.vq, nullptr, EQb, NN * 8, 64, 128);
    gemm(Xall, l.vk0, nullptr, EKb, NN * 3, 64, 128, 0, 3, 1024, 128, 0,   512, 64, 0);
    gemm(Xall, l.vk1, nullptr, EKb, NN * 5, 64, 128, 0, 5, 1024, 128, 384, 512, 64, 192);
    long tw = (long)NE * 64;
    k_htr_w<<<(unsigned)((tw + 255) / 256), 256, 0, stream>>>(EQb, EKb, n_j, n_i, wbuf, tw);
    lnorm(wbuf, wbuf, l.mw_lng, l.mw_lnb, 64, NE, 0);        // mlp_w pre-LN
    gemm(wbuf, l.mw_l1w, l.mw_l1b, w2, NE, 64, 64, 1);
    gemm(w2, l.mw_l2w, l.mw_l2b, EB1, NE, 128, 64);          // mw
    gemm(tij, l.mt_l1w, l.mt_l1b, EB3, NE, 128, 128, 1);
    gemm(EB3, l.mt_l2w, l.mt_l2b, EB2, NE, 128, 128);        // mt
    k_muladd<<<(unsigned)((128L * NE + 255) / 256), 256, 0, stream>>>(tij, EB1, EB2, 128L * NE);
    // ---- sea + combine ----
    run_sea(l.sea, l.rs_b, l.rs_w, l.ms_l1b, l.ms_l1w, l.ms_l2b, l.ms_l2w, hn, 640, 80);
    // ---- scatter updates ----
    zero(dX, 1024L * NN);
    k_scat_gata<<<NE, 128, 0, stream>>>(obuf, r1, shb, Xall, n_j, n_i, hn, dX);
    k_add<<<(unsigned)((1024L * NN + 255) / 256), 256, 0, stream>>>(Xall, dX, 1024L * NN);
    hother = hcur; hcur = hn;   // h <- hn (+ scattered parts[0])
    // ---- eqff ----
    gemm(Xall, l.vu_w, nullptr, Xvu, NN * 8, 128, 128);
    k_l2cat<<<NN, 128, 0, stream>>>(Xvu, hcur, catb);
    lnorm(catb, catln, l.mm_lng, l.mm_lnb, 256, NN, 0);
    gemm(catln, l.mm_l1w, l.mm_l1b, mh, NN, 128, 256, 1);
    gemm(mh, l.mm_l2w, l.mm_l2b, mo, NN, 256, 128);
    k_eqff<<<NN, 128, 0, stream>>>(mo, Xvu, hcur, Xall);
  }

  // ================== decoder ==================
  gemm(hcur, dec_l1_w, dec_l1_b, d1, NN, 128, 128, 1);
  zero((float*)d_out, 64);
  k_dec<<<(NN + 127) / 128, 128, 0, stream>>>(d1, dec_l2_w, dec_l2_b, batch, (float*)d_out);
}